// HGATLayer_63359357551441
// MI455X (gfx1250) — compile-verified
//
#include <hip/hip_runtime.h>
#include <math.h>

#define DIMF 128          // input feature dim
#define FDIM 256          // H*D output dim
#define EPSF 1e-15f
#define EPS2 1e-30f
#define MAXN 0.996f       // 1 - BALL_EPS

typedef float v2f __attribute__((ext_vector_type(2)));
typedef float v8f __attribute__((ext_vector_type(8)));

// ---------- device helpers ----------
__device__ __forceinline__ float wred32(float v) {
#pragma unroll
  for (int m = 16; m > 0; m >>= 1) v += __shfl_xor(v, m, 32);
  return v;
}
__device__ __forceinline__ float artanh_c(float x) {
  x = fminf(fmaxf(x, -1.0f + 1e-7f), 1.0f - 1e-7f);
  return 0.5f * logf((1.0f + x) / (1.0f - x));
}
__device__ __forceinline__ void atomicMaxF(float* addr, float v) {
  int* ai = (int*)addr;
  int old = __float_as_int(*addr);
  while (v > __int_as_float(old)) {
    int assumed = old;
    old = atomicCAS(ai, assumed, __float_as_int(v));
    if (old == assumed) break;
  }
}

// ---------- K4: fill ----------
__global__ void k_fill(float* p, float v, size_t n) {
  size_t i = (size_t)blockIdx.x * blockDim.x + threadIdx.x;
  if (i < n) p[i] = v;
}

// ---------- K1: time feature + projx + mobius_add(hyper, time_feat) ----------
__global__ void k_node_prep(const float* __restrict__ hyper,
                            const float* __restrict__ dt,
                            const float* __restrict__ tw,
                            const float* __restrict__ tb,
                            float* __restrict__ srcFeat,
                            float* __restrict__ xnArr,
                            int nTot, int nDst) {
  int node = (int)((blockIdx.x * (size_t)blockDim.x + threadIdx.x) >> 5);
  int lane = threadIdx.x & 31;
  if (node >= nTot) return;
  float t = (node < nDst) ? 0.0f : dt[node - nDst];
  float4 x  = ((const float4*)(hyper + (size_t)node * DIMF))[lane];
  float4 w4 = ((const float4*)tw)[lane];
  float4 b4 = ((const float4*)tb)[lane];
  float4 y;
  y.x = cosf(t * w4.x + b4.x); y.y = cosf(t * w4.y + b4.y);
  y.z = cosf(t * w4.z + b4.z); y.w = cosf(t * w4.w + b4.w);
  float y2raw = wred32(y.x*y.x + y.y*y.y + y.z*y.z + y.w*y.w);
  float n = sqrtf(fmaxf(y2raw, EPS2));
  float s = (n > MAXN) ? (MAXN / n) : 1.0f;
  y.x *= s; y.y *= s; y.z *= s; y.w *= s;
  float y2 = y2raw * s * s;
  float x2 = wred32(x.x*x.x + x.y*x.y + x.z*x.z + x.w*x.w);
  float xy = wred32(x.x*y.x + x.y*y.y + x.z*y.z + x.w*y.w);
  float den = fmaxf(1.0f + 2.0f*xy + x2*y2, EPSF);
  float c1 = 1.0f + 2.0f*xy + y2, c2 = 1.0f - x2;
  float4 o;
  o.x = (c1*x.x + c2*y.x) / den; o.y = (c1*x.y + c2*y.y) / den;
  o.z = (c1*x.z + c2*y.z) / den; o.w = (c1*x.w + c2*y.w) / den;
  ((float4*)(srcFeat + (size_t)node * DIMF))[lane] = o;
  float n2 = wred32(o.x*o.x + o.y*o.y + o.z*o.z + o.w*o.w);
  if (lane == 0) xnArr[node] = sqrtf(fmaxf(n2, EPS2));
}

// ---------- K2/K3: WMMA GEMM (X @ W^T) + mobius_matvec/mobius_add/logmap0 post ----------
__global__ __launch_bounds__(256)
void k_matvec_post(const float* __restrict__ X,      // N x 128
                   const float* __restrict__ xnArr,  // N
                   const float* __restrict__ W,      // 256 x 128 (row-major)
                   const float* __restrict__ bvec,   // 256
                   const float* __restrict__ attn,   // 256 (H*D)
                   float* __restrict__ feat,         // N x 256
                   float* __restrict__ lsArr,        // N  (logmap0 scale)
                   float* __restrict__ n2Arr,        // N  (|feat|^2 raw)
                   float* __restrict__ adots,        // N x 2 (feat . attn_h)
                   int N) {
  __shared__ float xs[16 * DIMF];    // 8 KB A strip
  __shared__ float tile[16 * FDIM];  // 16 KB mx tile
  const int rowBase = blockIdx.x * 16;
  const int tid  = threadIdx.x;
  const int w    = tid >> 5;
  const int lane = tid & 31;
  const int hlf  = lane >> 4;
  const int lm   = lane & 15;

  // stage 16x128 X strip into LDS (zero-pad tail rows)
  for (int g = tid; g < 16 * 32; g += 256) {
    int r = g >> 5, c4 = g & 31, gr = rowBase + r;
    float4 v = make_float4(0.f, 0.f, 0.f, 0.f);
    if (gr < N) v = ((const float4*)(X + (size_t)gr * DIMF))[c4];
    ((float4*)xs)[g] = v;
  }
  __syncthreads();

  // each wave: two 16x16 output tiles covering cols [32w, 32w+32)
  v8f acc0 = {}; v8f acc1 = {};
  const int col0 = w * 32 + lm;
  const int col1 = col0 + 16;
  for (int kb = 0; kb < DIMF; kb += 4) {
    int kk = kb + 2 * hlf;
    v2f a;  a.x = xs[lm * DIMF + kk];             a.y = xs[lm * DIMF + kk + 1];
    v2f b0; b0.x = W[(size_t)col0 * DIMF + kk];   b0.y = W[(size_t)col0 * DIMF + kk + 1];
    v2f b1; b1.x = W[(size_t)col1 * DIMF + kk];   b1.y = W[(size_t)col1 * DIMF + kk + 1];
    acc0 = __builtin_amdgcn_wmma_f32_16x16x4_f32(false, a, false, b0, (short)0, acc0, false, false);
    acc1 = __builtin_amdgcn_wmma_f32_16x16x4_f32(false, a, false, b1, (short)0, acc1, false, false);
  }
#pragma unroll
  for (int r = 0; r < 8; ++r) {
    tile[(r + 8*hlf) * FDIM + w*32      + lm] = acc0[r];
    tile[(r + 8*hlf) * FDIM + w*32 + 16 + lm] = acc1[r];
  }
  __syncthreads();

  // postprocess: wave w owns rows 2w and 2w+1 of the tile
  for (int rr = 0; rr < 2; ++rr) {
    int r = 2*w + rr, grow = rowBase + r;
    bool valid = grow < N;
    float xn = valid ? xnArr[grow] : 1.0f;
    const float* trow = tile + r * FDIM;
    float mloc[8], bloc[8];
    float mm = 0.f, mb = 0.f, bb = 0.f;
#pragma unroll
    for (int q = 0; q < 8; ++q) {
      int j = lane * 8 + q;
      float m = trow[j], bj = bvec[j];
      mloc[q] = m; bloc[q] = bj;
      mm += m * m; mb += m * bj; bb += bj * bj;
    }
    mm = wred32(mm); mb = wred32(mb); bb = wred32(bb);
    float mxn = sqrtf(fmaxf(mm, EPS2));
    float s1  = tanhf((mxn / xn) * artanh_c(xn)) / mxn;   // res = s1 * mx
    float x2 = s1 * s1 * mm, xy = s1 * mb, y2 = bb;
    float den = fmaxf(1.0f + 2.0f*xy + x2*y2, EPSF);
    float ca = 1.0f + 2.0f*xy + y2, cb = 1.0f - x2;
    float n2 = 0.f, aacc = 0.f;                            // h = (lane<16)?0:1
#pragma unroll
    for (int q = 0; q < 8; ++q) {
      int j = lane * 8 + q;
      float fj = (ca * (s1 * mloc[q]) + cb * bloc[q]) / den;
      if (valid) feat[(size_t)grow * FDIM + j] = fj;
      n2 += fj * fj;
      aacc += fj * attn[j];
    }
    n2 = wred32(n2);
#pragma unroll
    for (int m = 8; m > 0; m >>= 1) aacc += __shfl_xor(aacc, m, 32); // reduce within 16-lane halves
    float a1 = __shfl(aacc, 16, 32);
    if (valid && lane == 0) {
      float nrm = sqrtf(fmaxf(n2, EPS2));
      lsArr[grow] = artanh_c(nrm) / nrm;
      n2Arr[grow] = n2;
      adots[(size_t)grow * 2 + 0] = aacc;
      adots[(size_t)grow * 2 + 1] = a1;
    }
  }
}

// ---------- K5: per-edge ball_dist + raw attention logits + segmax(inv) ----------
__global__ void k_edge_dist(const float* __restrict__ featS, const float* __restrict__ featD,
                            const float* __restrict__ n2S,  const float* __restrict__ n2D,
                            const float* __restrict__ lsS,  const float* __restrict__ lsD,
                            const float* __restrict__ asA,  const float* __restrict__ adA,
                            const int* __restrict__ srcIdx, const int* __restrict__ dstIdx,
                            float* __restrict__ invArr, float* __restrict__ eRaw,
                            float* __restrict__ mInv, int NE) {
  int i = (int)((blockIdx.x * (size_t)blockDim.x + threadIdx.x) >> 5);
  int lane = threadIdx.x & 31;
  if (i >= NE) return;
  int s = srcIdx[i], t = dstIdx[i];
  const float4* fs = (const float4*)(featS + (size_t)s * FDIM);
  const float4* fd = (const float4*)(featD + (size_t)t * FDIM);
  float xy = 0.f;
#pragma unroll
  for (int q = 0; q < 2; ++q) {
    float4 a = fs[lane * 2 + q], b = fd[lane * 2 + q];
    xy += a.x*b.x + a.y*b.y + a.z*b.z + a.w*b.w;
  }
  xy = wred32(xy);
  float x2 = n2S[s], y2 = n2D[t];
  // |mobius_add(-x,y)|^2 = (A^2 x2 - 2AB xy + B^2 y2)/den^2 ; A=1-2xy+y2, B=1-x2
  float A = 1.0f - 2.0f*xy + y2, B = 1.0f - x2;
  float den = fmaxf(1.0f - 2.0f*xy + x2*y2, EPSF);
  float nn = A*A*x2 - 2.0f*A*B*xy + B*B*y2;
  float val = sqrtf(fmaxf(nn / (den * den), EPS2));
  float dd = 2.0f * artanh_c(val);
  float inv = 1.0f / (1e-15f + dd);
  if (lane == 0) {
    invArr[i] = inv;
    float l0 = lsS[s], l1 = lsD[t];
    eRaw[(size_t)i*2 + 0] = l0 * asA[(size_t)s*2 + 0] + l1 * adA[(size_t)t*2 + 0];
    eRaw[(size_t)i*2 + 1] = l0 * asA[(size_t)s*2 + 1] + l1 * adA[(size_t)t*2 + 1];
    atomicMaxF(&mInv[t], inv);
  }
}

// ---------- K6-K8: edge softmax passes ----------
__global__ void k_soft1(const float* invArr, const float* mInv, const int* dstIdx,
                        float* zArr, float* sZ, int NE) {
  int i = blockIdx.x * blockDim.x + threadIdx.x;
  if (i >= NE) return;
  int t = dstIdx[i];
  float z = expf(invArr[i] - mInv[t]);
  zArr[i] = z;
  atomicAdd(&sZ[t], z);
}
__global__ void k_soft2(const float* zArr, const float* sZ, const float* eRaw,
                        const int* dstIdx, float* e2, float* mE, int NE) {
  int idx = blockIdx.x * blockDim.x + threadIdx.x;
  if (idx >= 2 * NE) return;
  int i = idx >> 1, h = idx & 1, t = dstIdx[i];
  float dsm = zArr[i] / sZ[t];
  float v = eRaw[idx] * dsm;
  v = (v > 0.0f) ? v : 0.2f * v;     // leaky_relu(.,0.2)
  e2[idx] = v;
  atomicMaxF(&mE[(size_t)t*2 + h], v);
}
__global__ void k_soft3(const float* e2, const float* mE, const int* dstIdx,
                        float* zeArr, float* sE, int NE) {
  int idx = blockIdx.x * blockDim.x + threadIdx.x;
  if (idx >= 2 * NE) return;
  int i = idx >> 1, h = idx & 1, t = dstIdx[i];
  float ze = expf(e2[idx] - mE[(size_t)t*2 + h]);
  zeArr[idx] = ze;
  atomicAdd(&sE[(size_t)t*2 + h], ze);
}

// ---------- K9: ft[t] += el[s] * a  (el[s] = lsS[s]*featS[s]) ----------
__global__ void k_edge_accum(const float* __restrict__ featS, const float* __restrict__ lsS,
                             const int* __restrict__ srcIdx, const int* __restrict__ dstIdx,
                             const float* __restrict__ zeArr, const float* __restrict__ sE,
                             float* __restrict__ ft, int NE) {
  int i = (int)((blockIdx.x * (size_t)blockDim.x + threadIdx.x) >> 5);
  int lane = threadIdx.x & 31;
  if (i >= NE) return;
  int s = srcIdx[i], t = dstIdx[i];
  float l0 = lsS[s];
  float a0 = zeArr[(size_t)i*2 + 0] / sE[(size_t)t*2 + 0] * l0;
  float a1 = zeArr[(size_t)i*2 + 1] / sE[(size_t)t*2 + 1] * l0;
  float coeff = (lane < 16) ? a0 : a1;
  const float* fr = featS + (size_t)s * FDIM;
  float* fo = ft + (size_t)t * FDIM;
#pragma unroll
  for (int q = 0; q < 8; ++q) {
    int j = lane * 8 + q;
    atomicAdd(&fo[j], coeff * fr[j]);
  }
}

// ---------- K10: out = expmap0(ft) ----------
__global__ void k_expmap(const float* __restrict__ ft, float* __restrict__ out, int N) {
  int n = (int)((blockIdx.x * (size_t)blockDim.x + threadIdx.x) >> 5);
  int lane = threadIdx.x & 31;
  if (n >= N) return;
  const float4* u = (const float4*)(ft + (size_t)n * FDIM);
  float4 v0 = u[lane*2], v1 = u[lane*2 + 1];
  float n2 = v0.x*v0.x + v0.y*v0.y + v0.z*v0.z + v0.w*v0.w
           + v1.x*v1.x + v1.y*v1.y + v1.z*v1.z + v1.w*v1.w;
  n2 = wred32(n2);
  float un = sqrtf(fmaxf(n2, EPS2));
  float sc = tanhf(un) / un;
  float4* o = (float4*)(out + (size_t)n * FDIM);
  v0.x *= sc; v0.y *= sc; v0.z *= sc; v0.w *= sc;
  v1.x *= sc; v1.y *= sc; v1.z *= sc; v1.w *= sc;
  o[lane*2] = v0; o[lane*2 + 1] = v1;
}

// ---------- host launcher ----------
extern "C" void kernel_launch(void* const* d_in, const int* in_sizes, int n_in,
                              void* d_out, int out_size, void* d_ws, size_t ws_size,
                              hipStream_t stream) {
  const float* hyper  = (const float*)d_in[0];
  const float* dt     = (const float*)d_in[1];
  const float* time_w = (const float*)d_in[2];
  const float* time_b = (const float*)d_in[3];
  const float* w_src  = (const float*)d_in[4];
  const float* b_src  = (const float*)d_in[5];
  const float* w_dst  = (const float*)d_in[6];
  const float* b_dst  = (const float*)d_in[7];
  const float* attn   = (const float*)d_in[8];
  const int*   srcIdx = (const int*)d_in[9];
  const int*   dstIdx = (const int*)d_in[10];

  const int NE = in_sizes[1];                 // edges
  const int NT = in_sizes[0] / DIMF;          // total nodes
  const int ND = NT - NE;                     // dst nodes

  // workspace carve (fp32)
  char* base = (char*)d_ws;
  size_t off = 0;
  auto alloc = [&](size_t nFloats) {
    float* p = (float*)(base + off);
    off += ((nFloats * sizeof(float)) + 255) & ~(size_t)255;
    return p;
  };
  float* srcFeat = alloc((size_t)NT * DIMF);
  float* xnArr   = alloc(NT);
  float* featS   = alloc((size_t)NT * FDIM);
  float* lsS     = alloc(NT);
  float* n2S     = alloc(NT);
  float* asA     = alloc((size_t)NT * 2);
  float* featD   = alloc((size_t)ND * FDIM);
  float* lsD     = alloc(ND);
  float* n2D     = alloc(ND);
  float* adA     = alloc((size_t)ND * 2);
  float* invArr  = alloc(NE);
  float* eRaw    = alloc((size_t)NE * 2);
  float* zArr    = alloc(NE);
  float* e2Arr   = alloc((size_t)NE * 2);
  float* zeArr   = alloc((size_t)NE * 2);
  float* mInv    = alloc(ND);
  float* sZ      = alloc(ND);
  float* mE      = alloc((size_t)ND * 2);
  float* sE      = alloc((size_t)ND * 2);
  float* ft      = alloc((size_t)ND * FDIM);

  // K1: node prep (one wave per node)
  k_node_prep<<<(NT + 7) / 8, 256, 0, stream>>>(hyper, dt, time_w, time_b,
                                                srcFeat, xnArr, NT, ND);
  // K2/K3: WMMA GEMM + hyperbolic postprocess
  k_matvec_post<<<(NT + 15) / 16, 256, 0, stream>>>(srcFeat, xnArr, w_src, b_src, attn,
                                                    featS, lsS, n2S, asA, NT);
  k_matvec_post<<<(ND + 15) / 16, 256, 0, stream>>>(srcFeat, xnArr, w_dst, b_dst, attn,
                                                    featD, lsD, n2D, adA, ND);
  // K4: init segment buffers + ft accumulator
  k_fill<<<(ND + 255) / 256, 256, 0, stream>>>(mInv, 0.0f, (size_t)ND);
  k_fill<<<(ND + 255) / 256, 256, 0, stream>>>(sZ, 0.0f, (size_t)ND);
  k_fill<<<(2 * ND + 255) / 256, 256, 0, stream>>>(mE, -1e30f, (size_t)ND * 2);
  k_fill<<<(2 * ND + 255) / 256, 256, 0, stream>>>(sE, 0.0f, (size_t)ND * 2);
  k_fill<<<((size_t)ND * FDIM + 255) / 256, 256, 0, stream>>>(ft, 0.0f, (size_t)ND * FDIM);
  // K5: edge dist + logits + segmax(inv)
  k_edge_dist<<<(NE + 7) / 8, 256, 0, stream>>>(featS, featD, n2S, n2D, lsS, lsD, asA, adA,
                                                srcIdx, dstIdx, invArr, eRaw, mInv, NE);
  // K6-K8: two edge softmaxes
  k_soft1<<<(NE + 255) / 256, 256, 0, stream>>>(invArr, mInv, dstIdx, zArr, sZ, NE);
  k_soft2<<<(2 * NE + 255) / 256, 256, 0, stream>>>(zArr, sZ, eRaw, dstIdx, e2Arr, mE, NE);
  k_soft3<<<(2 * NE + 255) / 256, 256, 0, stream>>>(e2Arr, mE, dstIdx, zeArr, sE, NE);
  // K9: weighted scatter-add
  k_edge_accum<<<(NE + 7) / 8, 256, 0, stream>>>(featS, lsS, srcIdx, dstIdx, zeArr, sE, ft, NE);
  // K10: expmap0 -> output
  k_expmap<<<(ND + 7) / 8, 256, 0, stream>>>(ft, (float*)d_out, ND);
}